// MessagePassing_15058155340156
// MI455X (gfx1250) — compile-verified
//
#include <hip/hip_runtime.h>
#include <hip/hip_bf16.h>

// Problem constants (match reference)
#define N_NODES   50000
#define N_EDGES   800000
#define NODE_DIM  128
#define EDGE_DIM  64
#define T_ROUNDS  3
#define CAT_U     (2*NODE_DIM + EDGE_DIM)   // 320 (edge message input width)
#define CAT_M     (2*NODE_DIM)              // 256 (node update input width)

#define EPB   64                 // edges per block
#define NPB   64                 // nodes per block
#define ROW_U (CAT_U + 4)        // 324 floats: stride pad -> conflict-free b64 fragment reads
#define ROW_M (CAT_M + 4)        // 260 floats (260 mod 64 == 4, same property)

typedef __attribute__((ext_vector_type(2))) float v2f;
typedef __attribute__((ext_vector_type(8))) float v8f;

// D(16x16,f32) = A(16x4,f32) x B(4x16,f32) + C   (V_WMMA_F32_16X16X4_F32)
static __device__ __forceinline__ v8f wmma_f32(v2f a, v2f b, v8f c) {
    return __builtin_amdgcn_wmma_f32_16x16x4_f32(
        /*neg_a=*/false, a, /*neg_b=*/false, b,
        /*c_mod=*/(short)0, c, /*reuse_a=*/false, /*reuse_b=*/false);
}

// Async global->LDS 16B copy (GLOBAL_LOAD_ASYNC_TO_LDS_B128, ASYNCcnt-tracked).
// ldsOff: wave-relative LDS byte address (low 32 bits of generic LDS pointer).
static __device__ __forceinline__ void async_g2l_b128(unsigned ldsOff, const void* g) {
    unsigned long long ga = (unsigned long long)g;
    asm volatile("global_load_async_to_lds_b128 %0, %1, off"
                 :: "v"(ldsOff), "v"(ga) : "memory");
}
static __device__ __forceinline__ void wait_async0() {
    asm volatile("s_wait_asynccnt 0x0" ::: "memory");
}
static __device__ __forceinline__ unsigned lds_off(const void* p) {
    return (unsigned)(unsigned long long)p;   // low 32 bits = LDS offset
}

__global__ void zero_f32(float* __restrict__ p, int n) {
    int i = blockIdx.x * blockDim.x + threadIdx.x;
    if (i < n) p[i] = 0.0f;
}

// ---------------------------------------------------------------------------
// Edge kernel: msg[e] = U_k @ [h[src[e]], h[dst[e]], edge_feat[e]] + U_b
//              agg[src[e]] += msg[e]     (fp32 hardware atomics, L2-resident)
// LDS layout: [128][ROW_U] weights (async-staged) | [64][ROW_U] cat tile
// ---------------------------------------------------------------------------
__global__ __launch_bounds__(256)
void edge_msg_kernel(const float* __restrict__ h,
                     const float* __restrict__ ef,
                     const int*   __restrict__ src,
                     const int*   __restrict__ dst,
                     const float* __restrict__ Uw,   // [NODE_DIM][CAT_U]
                     const float* __restrict__ Ub,   // [NODE_DIM]
                     float*       __restrict__ agg)  // [N_NODES][NODE_DIM]
{
    extern __shared__ float lds[];
    float* wlds = lds;                       // [NODE_DIM][ROW_U]
    float* cat  = lds + NODE_DIM * ROW_U;    // [EPB][ROW_U]

    const int  tid = threadIdx.x;
    const long e0  = (long)blockIdx.x * EPB;

    // ---- async-stage weight matrix U_k: 128 rows x 80 float4 (coalesced) ----
    for (int w = tid; w < NODE_DIM * 80; w += 256) {
        const int o = w / 80, part = w % 80;
        async_g2l_b128(lds_off(wlds + o * ROW_U + part * 4),
                       (const float4*)Uw + (o * 80 + part));
    }
    // ---- async-stage gathered cat tile: 64 edges x 80 float4 ----
    for (int c = tid; c < EPB * 80; c += 256) {
        const int  e    = c / 80;
        const int  part = c % 80;
        const long ge   = e0 + e;
        const float4* ga;
        if (part < 32)       ga = (const float4*)(h + (long)src[ge] * NODE_DIM) + part;
        else if (part < 64)  ga = (const float4*)(h + (long)dst[ge] * NODE_DIM) + (part - 32);
        else                 ga = (const float4*)(ef + ge * EDGE_DIM) + (part - 64);
        async_g2l_b128(lds_off(cat + e * ROW_U + part * 4), ga);
    }
    wait_async0();
    __syncthreads();

    // ---- wave decomposition: 4 edge-groups x 2 output halves ----
    const int lane = tid & 31;
    const int wave = tid >> 5;
    const int eg   = wave & 3;     // edge group (16 edges)
    const int nh   = wave >> 2;    // output half (64 outs)
    const int nl   = lane & 15;    // A-row within group / B-column
    const int kh   = lane >> 4;    // K-half of the 16x16x4 fragment

    const float* catRow = cat + (eg * 16 + nl) * ROW_U + 2 * kh;
    const int o0 = (nh * 4 + 0) * 16 + nl;
    const int o1 = (nh * 4 + 1) * 16 + nl;
    const int o2 = (nh * 4 + 2) * 16 + nl;
    const int o3 = (nh * 4 + 3) * 16 + nl;
    const float* b0p = wlds + o0 * ROW_U + 2 * kh;
    const float* b1p = wlds + o1 * ROW_U + 2 * kh;
    const float* b2p = wlds + o2 * ROW_U + 2 * kh;
    const float* b3p = wlds + o3 * ROW_U + 2 * kh;

    v8f acc0 = {}, acc1 = {}, acc2 = {}, acc3 = {};
    #pragma unroll 4
    for (int k = 0; k < CAT_U; k += 4) {
        const v2f a = *(const v2f*)(catRow + k);
        acc0 = wmma_f32(a, *(const v2f*)(b0p + k), acc0);
        acc1 = wmma_f32(a, *(const v2f*)(b1p + k), acc1);
        acc2 = wmma_f32(a, *(const v2f*)(b2p + k), acc2);
        acc3 = wmma_f32(a, *(const v2f*)(b3p + k), acc3);
    }

    // ---- epilogue: + bias (per message, summed like the reference), scatter ----
    const float bias0 = Ub[o0], bias1 = Ub[o1], bias2 = Ub[o2], bias3 = Ub[o3];
    const long eb = e0 + eg * 16;
    #pragma unroll
    for (int r = 0; r < 8; ++r) {
        const int  m    = r + 8 * kh;             // edge row per C/D layout
        const int  node = src[eb + m];
        float* aggRow   = agg + (long)node * NODE_DIM;
        unsafeAtomicAdd(aggRow + o0, acc0[r] + bias0);
        unsafeAtomicAdd(aggRow + o1, acc1[r] + bias1);
        unsafeAtomicAdd(aggRow + o2, acc2[r] + bias2);
        unsafeAtomicAdd(aggRow + o3, acc3[r] + bias3);
    }
}

// ---------------------------------------------------------------------------
// Node kernel: h[v] = relu(M_k @ [h[v], agg[v]] + M_b)   (in-place safe)
// LDS layout: [128][ROW_M] weights (async-staged) | [64][ROW_M] cat tile
// ---------------------------------------------------------------------------
__global__ __launch_bounds__(256)
void node_update_kernel(float*       __restrict__ h,
                        const float* __restrict__ agg,
                        const float* __restrict__ Mw,  // [NODE_DIM][CAT_M]
                        const float* __restrict__ Mb,  // [NODE_DIM]
                        int n_nodes)
{
    extern __shared__ float lds[];
    float* wlds = lds;                       // [NODE_DIM][ROW_M]
    float* cat  = lds + NODE_DIM * ROW_M;    // [NPB][ROW_M]

    const int  tid = threadIdx.x;
    const long v0  = (long)blockIdx.x * NPB;

    // ---- async-stage weight matrix M_k: 128 rows x 64 float4 ----
    for (int w = tid; w < NODE_DIM * 64; w += 256) {
        const int o = w / 64, part = w % 64;
        async_g2l_b128(lds_off(wlds + o * ROW_M + part * 4),
                       (const float4*)Mw + (o * 64 + part));
    }
    // ---- stage cat tile (tail-guarded: zero-fill past n_nodes) ----
    for (int c = tid; c < NPB * 64; c += 256) {
        const int  vl   = c / 64;
        const int  part = c % 64;
        const long v    = v0 + vl;
        float4 val = {0.f, 0.f, 0.f, 0.f};
        if (v < n_nodes) {
            if (part < 32) val = ((const float4*)(h   + v * NODE_DIM))[part];
            else           val = ((const float4*)(agg + v * NODE_DIM))[part - 32];
        }
        *(float4*)(cat + vl * ROW_M + part * 4) = val;
    }
    wait_async0();
    __syncthreads();

    const int lane = tid & 31;
    const int wave = tid >> 5;
    const int vg   = wave & 3;
    const int nh   = wave >> 2;
    const int nl   = lane & 15;
    const int kh   = lane >> 4;

    const float* catRow = cat + (vg * 16 + nl) * ROW_M + 2 * kh;
    const int o0 = (nh * 4 + 0) * 16 + nl;
    const int o1 = (nh * 4 + 1) * 16 + nl;
    const int o2 = (nh * 4 + 2) * 16 + nl;
    const int o3 = (nh * 4 + 3) * 16 + nl;
    const float* b0p = wlds + o0 * ROW_M + 2 * kh;
    const float* b1p = wlds + o1 * ROW_M + 2 * kh;
    const float* b2p = wlds + o2 * ROW_M + 2 * kh;
    const float* b3p = wlds + o3 * ROW_M + 2 * kh;

    v8f acc0 = {}, acc1 = {}, acc2 = {}, acc3 = {};
    #pragma unroll 4
    for (int k = 0; k < CAT_M; k += 4) {
        const v2f a = *(const v2f*)(catRow + k);
        acc0 = wmma_f32(a, *(const v2f*)(b0p + k), acc0);
        acc1 = wmma_f32(a, *(const v2f*)(b1p + k), acc1);
        acc2 = wmma_f32(a, *(const v2f*)(b2p + k), acc2);
        acc3 = wmma_f32(a, *(const v2f*)(b3p + k), acc3);
    }

    const float bias0 = Mb[o0], bias1 = Mb[o1], bias2 = Mb[o2], bias3 = Mb[o3];
    #pragma unroll
    for (int r = 0; r < 8; ++r) {
        const long v = v0 + vg * 16 + r + 8 * kh;
        if (v < n_nodes) {
            float* hRow = h + v * NODE_DIM;
            hRow[o0] = fmaxf(acc0[r] + bias0, 0.0f);
            hRow[o1] = fmaxf(acc1[r] + bias1, 0.0f);
            hRow[o2] = fmaxf(acc2[r] + bias2, 0.0f);
            hRow[o3] = fmaxf(acc3[r] + bias3, 0.0f);
        }
    }
}

extern "C" void kernel_launch(void* const* d_in, const int* in_sizes, int n_in,
                              void* d_out, int out_size, void* d_ws, size_t ws_size,
                              hipStream_t stream) {
    const float* node_feat = (const float*)d_in[0];
    const float* edge_feat = (const float*)d_in[1];
    const int*   src       = (const int*)d_in[2];
    const int*   dst       = (const int*)d_in[3];
    const float* U_w       = (const float*)d_in[4];  // [T][128][320]
    const float* U_b       = (const float*)d_in[5];  // [T][128]
    const float* M_w       = (const float*)d_in[6];  // [T][128][256]
    const float* M_b       = (const float*)d_in[7];  // [T][128]

    float* h   = (float*)d_ws;                                 // 25.6 MB
    float* agg = h + (size_t)N_NODES * NODE_DIM;               // 25.6 MB

    hipMemcpyAsync(h, node_feat, sizeof(float) * (size_t)N_NODES * NODE_DIM,
                   hipMemcpyDeviceToDevice, stream);

    const size_t ldsE = (size_t)(NODE_DIM + EPB) * ROW_U * sizeof(float); // ~249 KB
    const size_t ldsN = (size_t)(NODE_DIM + NPB) * ROW_M * sizeof(float); // ~200 KB
    const int aggN = N_NODES * NODE_DIM;

    for (int k = 0; k < T_ROUNDS; ++k) {
        zero_f32<<<(aggN + 255) / 256, 256, 0, stream>>>(agg, aggN);
        edge_msg_kernel<<<N_EDGES / EPB, 256, ldsE, stream>>>(
            h, edge_feat, src, dst,
            U_w + (size_t)k * NODE_DIM * CAT_U,
            U_b + (size_t)k * NODE_DIM, agg);
        node_update_kernel<<<(N_NODES + NPB - 1) / NPB, 256, ldsN, stream>>>(
            h, agg,
            M_w + (size_t)k * NODE_DIM * CAT_M,
            M_b + (size_t)k * NODE_DIM, N_NODES);
    }

    hipMemcpyAsync(d_out, edge_feat, sizeof(float) * (size_t)N_EDGES * EDGE_DIM,
                   hipMemcpyDeviceToDevice, stream);
    hipMemcpyAsync((float*)d_out + (size_t)N_EDGES * EDGE_DIM, h,
                   sizeof(float) * (size_t)N_NODES * NODE_DIM,
                   hipMemcpyDeviceToDevice, stream);
}